// Net_38740605010536
// MI455X (gfx1250) — compile-verified
//
#include <hip/hip_runtime.h>
#include <hip/hip_bf16.h>
#include <math.h>

typedef __attribute__((ext_vector_type(2))) float v2f;
typedef __attribute__((ext_vector_type(8))) float v8f;

// ======================================================================
// WMMA f32 GEMM with fused GCN self-loop epilogue.
//   lin[i, n0+j]     = op(A) @ B                  (raw product, fed to edge_agg)
//   selfout[i, n0+j] = lin[i, n0+j]*dinv[i]^2 + bias[n0+j]
// op = identity or ReLU (fused on A load).
// One wave per 16-row strip, covering TN consecutive 16-col tiles starting
// at column n0; the A fragment is loaded once per K-step and reused for all
// TN WMMAs (8x less A traffic than one-tile-per-wave for TN=8).
//
// V_WMMA_F32_16X16X4_F32 fragment layouts (wave32):
//   A (16x4): lane l holds A[l&15][2*(l>>4) + {0,1}] in 2 VGPRs.
//   B (4x16): lane l holds B[2*(l>>4) + {0,1}][l&15].
//   C/D (16x16): vgpr g, lane l -> (row = g + 8*(l>>4), col = l&15).
// Requires M % 16 == 0, K % 4 == 0 (holds: M=50000, K in {256,128}).
// ======================================================================
template <bool RELU, int TN>
__global__ void gemm_wmma_f32(const float* __restrict__ A,
                              const float* __restrict__ B,
                              float* __restrict__ lin,
                              float* __restrict__ selfout,
                              const float* __restrict__ dinv,
                              const float* __restrict__ bias,
                              int M, int K, int N, int n0) {
  const int lane = threadIdx.x & 31;
  const int wavesPerBlock = blockDim.x >> 5;
  const int wave = blockIdx.x * wavesPerBlock + (threadIdx.x >> 5);
  const int tilesM = M >> 4;
  if (wave >= tilesM) return;  // wave-uniform: EXEC stays all-ones
  const int tm = wave;

  const int mrow = lane & 15;   // A row / B col / D col within tile
  const int kh   = lane >> 4;   // which K half-pair this lane owns
  const float* arow = A + (size_t)(tm * 16 + mrow) * K;

  v8f acc[TN] = {};
  for (int k = 0; k < K; k += 4) {
    const int ka = k + 2 * kh;
    float2 av = *(const float2*)(arow + ka);  // 8B aligned: K even, ka even
    if (RELU) { av.x = fmaxf(av.x, 0.0f); av.y = fmaxf(av.y, 0.0f); }
    v2f a; a.x = av.x; a.y = av.y;
    const float* brow0 = B + (size_t)(ka + 0) * N + n0 + mrow;
    const float* brow1 = B + (size_t)(ka + 1) * N + n0 + mrow;
#pragma unroll
    for (int t = 0; t < TN; ++t) {
      v2f b; b.x = brow0[t * 16]; b.y = brow1[t * 16];
      acc[t] = __builtin_amdgcn_wmma_f32_16x16x4_f32(
          /*neg_a=*/false, a, /*neg_b=*/false, b,
          /*c_mod=*/(short)0, acc[t], /*reuse_a=*/false, /*reuse_b=*/false);
    }
  }

#pragma unroll
  for (int t = 0; t < TN; ++t) {
    const int col = n0 + t * 16 + mrow;
    const float bv = bias[col];
    float* lp = lin + (size_t)(tm * 16) * N + col;
    float* sp = selfout + (size_t)(tm * 16) * N + col;
#pragma unroll
    for (int g = 0; g < 8; ++g) {
      const int r = g + 8 * kh;
      const float d = dinv[tm * 16 + r];
      const float v = acc[t][g];
      lp[(size_t)r * N] = v;
      sp[(size_t)r * N] = v * d * d + bv;
    }
  }
}

// ======================================================================
// Small utility kernels
// ======================================================================
__global__ void fill_f32(float* __restrict__ p, float v, int n) {
  int i = blockIdx.x * blockDim.x + threadIdx.x;
  if (i < n) p[i] = v;
}

// deg[col[e]] += 1  (original edges, weight 1)
__global__ void deg_add_edges(float* __restrict__ deg, const int* __restrict__ col,
                              int nE) {
  int e = blockIdx.x * blockDim.x + threadIdx.x;
  if (e < nE) atomicAdd(&deg[col[e]], 1.0f);
}

// Candidate edge e contributes weight m[e] to both deg[t1[e]] and deg[t0[e]]
// (augmented graph has the edge in both directions).
__global__ void deg_add_cand(float* __restrict__ deg, const int* __restrict__ t0,
                             const int* __restrict__ t1,
                             const float* __restrict__ m, int nc) {
  int e = blockIdx.x * blockDim.x + threadIdx.x;
  if (e < nc) {
    float w = m[e];
    if (w != 0.0f) {
      atomicAdd(&deg[t1[e]], w);
      atomicAdd(&deg[t0[e]], w);
    }
  }
}

__global__ void rsqrt_inplace(float* __restrict__ p, int n) {
  int i = blockIdx.x * blockDim.x + threadIdx.x;
  if (i < n) p[i] = 1.0f / sqrtf(p[i]);
}

// out[col[e], f] += lin[row[e], f] * dinv[row]*dinv[col]   (weight-1 edges)
// blockDim.x = 128; 128/F logical edges per block.
__global__ void edge_agg(const float* __restrict__ lin, float* __restrict__ out,
                         const int* __restrict__ rows, const int* __restrict__ cols,
                         const float* __restrict__ dinv, int nE, int F) {
  int epb = blockDim.x / F;
  int le = threadIdx.x / F;
  int f = threadIdx.x - le * F;
  int e = blockIdx.x * epb + le;
  if (e >= nE) return;
  int r = rows[e], c = cols[e];
  float norm = dinv[r] * dinv[c];
  atomicAdd(&out[(size_t)c * F + f], lin[(size_t)r * F + f] * norm);
}

// Both directions of candidate edge e with weight m[e].
__global__ void cand_agg(const float* __restrict__ lin, float* __restrict__ out,
                         const int* __restrict__ t0, const int* __restrict__ t1,
                         const float* __restrict__ m,
                         const float* __restrict__ dinv, int nc, int F) {
  int epb = blockDim.x / F;
  int le = threadIdx.x / F;
  int f = threadIdx.x - le * F;
  int e = blockIdx.x * epb + le;
  if (e >= nc) return;
  float w = m[e];
  if (w == 0.0f) return;
  int a = t0[e], b = t1[e];
  float norm = dinv[a] * w * dinv[b];
  atomicAdd(&out[(size_t)b * F + f], lin[(size_t)a * F + f] * norm);
  atomicAdd(&out[(size_t)a * F + f], lin[(size_t)b * F + f] * norm);
}

// One wave32 per candidate edge: m[e] = (dot(z[s], z[d]) > 0) since
// sigmoid(x) > 0.5 <=> x > 0. Accumulates accepted-edge count.
__global__ void decoder_k(const float* __restrict__ z, const int* __restrict__ s,
                          const int* __restrict__ d, float* __restrict__ m,
                          float* __restrict__ rcnt, int nc, int H) {
  int lane = threadIdx.x & 31;
  int e = blockIdx.x * (blockDim.x >> 5) + (threadIdx.x >> 5);
  if (e >= nc) return;
  const float* za = z + (size_t)s[e] * H;
  const float* zb = z + (size_t)d[e] * H;
  float acc = 0.0f;
  for (int f = lane; f < H; f += 32) acc += za[f] * zb[f];
#pragma unroll
  for (int off = 16; off > 0; off >>= 1) acc += __shfl_xor(acc, off, 32);
  if (lane == 0) {
    float mm = (acc > 0.0f) ? 1.0f : 0.0f;
    m[e] = mm;
    if (mm != 0.0f) atomicAdd(rcnt, 1.0f);
  }
}

// In-place row-wise log-softmax, one thread per row (C = 16).
__global__ void log_softmax_k(float* __restrict__ out, int n, int C) {
  int i = blockIdx.x * blockDim.x + threadIdx.x;
  if (i >= n) return;
  float* r = out + (size_t)i * C;
  float mx = r[0];
  for (int c = 1; c < C; ++c) mx = fmaxf(mx, r[c]);
  float s = 0.0f;
  for (int c = 0; c < C; ++c) s += expf(r[c] - mx);
  float lse = mx + logf(s);
  for (int c = 0; c < C; ++c) r[c] = r[c] - lse;
}

__global__ void write_scalar(float* __restrict__ dst, const float* __restrict__ src) {
  if (threadIdx.x == 0 && blockIdx.x == 0) dst[0] = src[0];
}

// ======================================================================
// Host-side orchestration (all launches on `stream`, graph-capture safe)
// ======================================================================
static inline int cdiv_i(int a, int b) { return (a + b - 1) / b; }

// Dispatch helper: pick TN=8 (H=128) or TN=1, falling back to per-16-col
// launches for unexpected widths.
template <bool RELU>
static void launch_gemm(const float* A, const float* B, float* lin, float* selfout,
                        const float* dinv, const float* bias,
                        int M, int K, int N, hipStream_t stream) {
  const int tilesM = M / 16;
  const int blocks = cdiv_i(tilesM, 4);
  if (N == 128) {
    gemm_wmma_f32<RELU, 8><<<blocks, 128, 0, stream>>>(A, B, lin, selfout, dinv,
                                                       bias, M, K, N, 0);
  } else if (N == 16) {
    gemm_wmma_f32<RELU, 1><<<blocks, 128, 0, stream>>>(A, B, lin, selfout, dinv,
                                                       bias, M, K, N, 0);
  } else {
    for (int n0 = 0; n0 < N; n0 += 16) {
      gemm_wmma_f32<RELU, 1><<<blocks, 128, 0, stream>>>(A, B, lin, selfout, dinv,
                                                         bias, M, K, N, n0);
    }
  }
}

extern "C" void kernel_launch(void* const* d_in, const int* in_sizes, int n_in,
                              void* d_out, int out_size, void* d_ws, size_t ws_size,
                              hipStream_t stream) {
  const float* x   = (const float*)d_in[0];
  const int*   pei = (const int*)d_in[1];
  const int*   nei = (const int*)d_in[2];
  const int*   ei  = (const int*)d_in[3];
  const float* W1  = (const float*)d_in[4];
  const float* b1  = (const float*)d_in[5];
  const float* W2  = (const float*)d_in[6];
  const float* b2  = (const float*)d_in[7];
  const float* W1s = (const float*)d_in[8];
  const float* b1s = (const float*)d_in[9];
  const float* W2s = (const float*)d_in[10];
  const float* b2s = (const float*)d_in[11];

  const int H    = in_sizes[5];         // 128
  const int C    = in_sizes[11];        // 16
  const int Fin  = in_sizes[4] / H;     // 256
  const int N    = in_sizes[0] / Fin;   // 50000
  const int E    = in_sizes[3] / 2;     // 800000
  const int EP   = in_sizes[1] / 2;     // 100000

  // workspace layout (floats)
  float* lin    = (float*)d_ws;              // [N*H]  raw GEMM output
  float* hbuf   = lin    + (size_t)N * H;    // [N*H]  aggregated h1 / h2
  float* dinv1  = hbuf   + (size_t)N * H;    // [N]
  float* dinv2  = dinv1  + N;                // [N]
  float* mbuf   = dinv2  + N;                // [2*EP] accepted-edge mask
  float* linout = mbuf   + 2 * EP;           // [N*C]  final raw GEMM output
  float* rcnt   = linout + (size_t)N * C;    // [1]

  float* ls   = (float*)d_out;               // [N, C] log-softmax output
  float* zout = ls + (size_t)N * C;          // [N, H] z output

  const int* row = ei;        const int* col = ei + E;
  const int* p0  = pei;       const int* p1  = pei + EP;
  const int* n0  = nei;       const int* n1  = nei + EP;

  // ---------------- encoder graph: deg^{-1/2} ----------------
  fill_f32<<<cdiv_i(N, 256), 256, 0, stream>>>(dinv1, 1.0f, N);
  deg_add_edges<<<cdiv_i(E, 256), 256, 0, stream>>>(dinv1, col, E);
  rsqrt_inplace<<<cdiv_i(N, 256), 256, 0, stream>>>(dinv1, N);

  // ---------------- encoder layer 1: h1 = GCN(x, W1, b1) ----------------
  // GEMM writes lin = x@W1 and hbuf = lin*dinv1^2 + b1 (self-loop init)
  launch_gemm<false>(x, W1, lin, hbuf, dinv1, b1, N, Fin, H, stream);
  edge_agg<<<cdiv_i(E * H, 128), 128, 0, stream>>>(lin, hbuf, row, col, dinv1, E, H);

  // ---------------- encoder layer 2: z = GCN(relu(h1), W2, b2) ----------------
  launch_gemm<true>(hbuf, W2, lin, zout, dinv1, b2, N, H, H, stream);
  edge_agg<<<cdiv_i(E * H, 128), 128, 0, stream>>>(lin, zout, row, col, dinv1, E, H);

  // ---------------- decoder: mask + accepted-edge count ----------------
  fill_f32<<<1, 32, 0, stream>>>(rcnt, 0.0f, 1);
  decoder_k<<<cdiv_i(EP, 4), 128, 0, stream>>>(zout, p0, p1, mbuf, rcnt, EP, H);
  decoder_k<<<cdiv_i(EP, 4), 128, 0, stream>>>(zout, n0, n1, mbuf + EP, rcnt, EP, H);

  // ---------------- augmented graph degrees ----------------
  fill_f32<<<cdiv_i(N, 256), 256, 0, stream>>>(dinv2, 1.0f, N);
  deg_add_edges<<<cdiv_i(E, 256), 256, 0, stream>>>(dinv2, col, E);
  deg_add_cand<<<cdiv_i(EP, 256), 256, 0, stream>>>(dinv2, p0, p1, mbuf, EP);
  deg_add_cand<<<cdiv_i(EP, 256), 256, 0, stream>>>(dinv2, n0, n1, mbuf + EP, EP);
  rsqrt_inplace<<<cdiv_i(N, 256), 256, 0, stream>>>(dinv2, N);

  // ---------------- second stack layer 1: h2 = GCN(x, W1s, b1s; aug) --------
  launch_gemm<false>(x, W1s, lin, hbuf, dinv2, b1s, N, Fin, H, stream);
  edge_agg<<<cdiv_i(E * H, 128), 128, 0, stream>>>(lin, hbuf, row, col, dinv2, E, H);
  cand_agg<<<cdiv_i(EP * H, 128), 128, 0, stream>>>(lin, hbuf, p0, p1, mbuf, dinv2, EP, H);
  cand_agg<<<cdiv_i(EP * H, 128), 128, 0, stream>>>(lin, hbuf, n0, n1, mbuf + EP, dinv2, EP, H);

  // ---------------- second stack layer 2: out = GCN(relu(h2), W2s, b2s; aug) -
  launch_gemm<true>(hbuf, W2s, linout, ls, dinv2, b2s, N, H, C, stream);
  edge_agg<<<cdiv_i(E * C, 128), 128, 0, stream>>>(linout, ls, row, col, dinv2, E, C);
  cand_agg<<<cdiv_i(EP * C, 128), 128, 0, stream>>>(linout, ls, p0, p1, mbuf, dinv2, EP, C);
  cand_agg<<<cdiv_i(EP * C, 128), 128, 0, stream>>>(linout, ls, n0, n1, mbuf + EP, dinv2, EP, C);

  // ---------------- log-softmax + r_size ----------------
  log_softmax_k<<<cdiv_i(N, 256), 256, 0, stream>>>(ls, N, C);
  write_scalar<<<1, 32, 0, stream>>>((float*)d_out + (out_size - 1), rcnt);
}